// GNNLayer_11441792876554
// MI455X (gfx1250) — compile-verified
//
#include <hip/hip_runtime.h>
#include <hip/hip_bf16.h>

// ---------------------------------------------------------------------------
// GAT layer for MI455X (gfx1250, wave32).
//   xl = x @ W^T          -> fp32 WMMA 16x16x4 (one 16x16 tile per wave)
//   ai/aj per node        -> wave-per-node dot products
//   segment softmax       -> ordered-uint atomicMax + float atomicAdd
//   scatter out[dst] += coef * xl[src]  -> wave-per-edge, 2 ch/lane atomics
//   BatchNorm + ReLU      -> per-channel sum/sumsq reduce, in-place apply
// ---------------------------------------------------------------------------

typedef __attribute__((ext_vector_type(2))) float v2f;
typedef __attribute__((ext_vector_type(8))) float v8f;

#define FCH 64           // F_IN == C == 64
#define NEG_SLOPE 0.2f

// ---------------- GEMM: xl[n][c] = sum_k x[n][k] * W[c][k] -----------------
// One wave computes one 16(rows) x 16(channels) tile. Block = 256 = 8 waves:
// wave w -> rowTile = 2*blockIdx + (w>>2), cTile = w&3 (C=64 -> 4 col tiles).
// fp32 WMMA 16x16x4: A is 16x4 (x tile), B is 4x16 (W^T tile).
// A layout (ISA 7.12.2): lane = 16*h + r holds A[r][2h + 4k .. 2h+1 + 4k].
// B layout mirrors:      lane = 16*h + r holds B[2h+4k..][r] = W[r'][2h+4k..]
// -> both fragments are float2 loads with identical index math.
__global__ void gemm_xl_kernel(const float* __restrict__ x,
                               const float* __restrict__ W,
                               float* __restrict__ xl, int n) {
    const int wave = threadIdx.x >> 5;
    const int lane = threadIdx.x & 31;
    const int rowBase = (blockIdx.x * 2 + (wave >> 2)) * 16;
    const int cBase   = (wave & 3) * 16;
    if (rowBase >= n) return;                 // N is a multiple of 16 here
    const int r = lane & 15;
    const int h = lane >> 4;

    const float* aPtr = x + (size_t)(rowBase + r) * FCH + 2 * h;
    const float* bPtr = W + (size_t)(cBase   + r) * FCH + 2 * h;

    v8f acc = {};
#pragma unroll
    for (int kk = 0; kk < 16; ++kk) {
        v2f a = *(const v2f*)(aPtr + 4 * kk);
        v2f b = *(const v2f*)(bPtr + 4 * kk);
        // 8 args: (neg_a, A, neg_b, B, c_mod, C, reuse_a, reuse_b)
        acc = __builtin_amdgcn_wmma_f32_16x16x4_f32(
            false, a, false, b, (short)0, acc, false, false);
    }
    // D layout: VGPR i -> row rowBase + i + 8*h, col cBase + r
    float* outp = xl + (size_t)(rowBase + 8 * h) * FCH + cBase + r;
#pragma unroll
    for (int i = 0; i < 8; ++i) outp[(size_t)i * FCH] = acc[i];
}

// ---------------- per-node attention coefficients ai / aj ------------------
__global__ void node_coef_kernel(const float* __restrict__ xl,
                                 const float* __restrict__ emb,
                                 const float* __restrict__ att_i,
                                 const float* __restrict__ att_j,
                                 const float* __restrict__ att_em_i,
                                 const float* __restrict__ att_em_j,
                                 float* __restrict__ ai,
                                 float* __restrict__ aj, int n) {
    const int node = (blockIdx.x * blockDim.x + threadIdx.x) >> 5;
    const int lane = threadIdx.x & 31;
    if (node >= n) return;
    const float* xr = xl  + (size_t)node * FCH;
    const float* er = emb + (size_t)node * FCH;
    float x0 = xr[lane], x1 = xr[lane + 32];
    float e0 = er[lane], e1 = er[lane + 32];
    float si = x0 * att_i[lane] + x1 * att_i[lane + 32]
             + e0 * att_em_i[lane] + e1 * att_em_i[lane + 32];
    float sj = x0 * att_j[lane] + x1 * att_j[lane + 32]
             + e0 * att_em_j[lane] + e1 * att_em_j[lane + 32];
#pragma unroll
    for (int off = 16; off; off >>= 1) {
        si += __shfl_down(si, off, 32);
        sj += __shfl_down(sj, off, 32);
    }
    if (lane == 0) { ai[node] = si; aj[node] = sj; }
}

// ---------------- edge helpers ---------------------------------------------
__device__ __forceinline__ void edge_ends(const long long* __restrict__ ei,
                                          int e, int E, int& s, int& d) {
    if (e < E) { s = (int)ei[e]; d = (int)ei[(size_t)E + e]; }
    else       { s = e - E; d = e - E; }     // appended self loops
}

__device__ __forceinline__ float lrelu(float a) {
    return a >= 0.0f ? a : NEG_SLOPE * a;
}

// monotonic float<->uint keys for atomicMax over signed floats
__device__ __forceinline__ unsigned enc_key(float f) {
    unsigned b = __float_as_uint(f);
    return (b & 0x80000000u) ? ~b : (b | 0x80000000u);
}
__device__ __forceinline__ float dec_key(unsigned k) {
    unsigned b = (k & 0x80000000u) ? (k & 0x7FFFFFFFu) : ~k;
    return __uint_as_float(b);
}

// ---------------- softmax pass 1: segment max over dst ---------------------
__global__ void edge_max_kernel(const long long* __restrict__ ei,
                                const float* __restrict__ ai,
                                const float* __restrict__ aj,
                                unsigned* __restrict__ mmax, int E, int n) {
    const int e = blockIdx.x * blockDim.x + threadIdx.x;
    if (e >= E + n) return;
    int s, d; edge_ends(ei, e, E, s, d);
    float a = lrelu(ai[d] + aj[s]);
    atomicMax(mmax + d, enc_key(a));
}

// ---------------- softmax pass 2: segment sum of exp -----------------------
__global__ void edge_sum_kernel(const long long* __restrict__ ei,
                                const float* __restrict__ ai,
                                const float* __restrict__ aj,
                                const unsigned* __restrict__ mmax,
                                float* __restrict__ ssum, int E, int n) {
    const int e = blockIdx.x * blockDim.x + threadIdx.x;
    if (e >= E + n) return;
    int s, d; edge_ends(ei, e, E, s, d);
    float a = lrelu(ai[d] + aj[s]);
    float ex = __expf(a - dec_key(mmax[d]));
    atomicAdd(ssum + d, ex);
}

// ---------------- message scatter: out[dst] += coef * xl[src] --------------
// One wave per edge; lane handles channels lane and lane+32.
__global__ void scatter_kernel(const long long* __restrict__ ei,
                               const float* __restrict__ xl,
                               const float* __restrict__ ai,
                               const float* __restrict__ aj,
                               const unsigned* __restrict__ mmax,
                               const float* __restrict__ ssum,
                               float* __restrict__ out, int E, int n) {
    const int e = blockIdx.x * (blockDim.x >> 5) + (threadIdx.x >> 5);
    const int lane = threadIdx.x & 31;
    if (e >= E + n) return;
    int s, d; edge_ends(ei, e, E, s, d);
    float a = lrelu(ai[d] + aj[s]);
    float coef = __expf(a - dec_key(mmax[d])) / (ssum[d] + 1e-16f);
    const float* xr = xl + (size_t)s * FCH;
    float* orow = out + (size_t)d * FCH;
    atomicAdd(orow + lane,      xr[lane]      * coef);
    atomicAdd(orow + lane + 32, xr[lane + 32] * coef);
}

// ---------------- BN stats: per-channel sum / sumsq of (out + bias) --------
__global__ void bn_stats_kernel(const float* __restrict__ out,
                                const float* __restrict__ bias,
                                float* __restrict__ sums,
                                float* __restrict__ sums2, int n) {
    const int c   = threadIdx.x & 63;
    const int sub = threadIdx.x >> 6;           // 4 row-slices per block
    const float b = bias[c];
    float s = 0.0f, s2 = 0.0f;
    for (int row = blockIdx.x * 4 + sub; row < n; row += gridDim.x * 4) {
        float v = out[(size_t)row * FCH + c] + b;
        s += v; s2 += v * v;
    }
    __shared__ float ls[256], ls2[256];
    ls[threadIdx.x] = s; ls2[threadIdx.x] = s2;
    __syncthreads();
    if (sub == 0) {
        s  = ls[c]  + ls[c + 64]  + ls[c + 128]  + ls[c + 192];
        s2 = ls2[c] + ls2[c + 64] + ls2[c + 128] + ls2[c + 192];
        atomicAdd(sums  + c, s);
        atomicAdd(sums2 + c, s2);
    }
}

// ---------------- BN apply + ReLU (in place on d_out) ----------------------
__global__ void bn_apply_kernel(float* __restrict__ out,
                                const float* __restrict__ bias,
                                const float* __restrict__ gamma,
                                const float* __restrict__ beta,
                                const float* __restrict__ sums,
                                const float* __restrict__ sums2, int n) {
    const size_t i = (size_t)blockIdx.x * blockDim.x + threadIdx.x;
    if (i >= (size_t)n * FCH) return;
    const int c = (int)(i & 63);
    const float invN = 1.0f / (float)n;
    const float mu  = sums[c] * invN;
    const float var = sums2[c] * invN - mu * mu;
    const float v = out[i] + bias[c];
    const float r = gamma[c] * (v - mu) * rsqrtf(var + 1e-5f) + beta[c];
    out[i] = fmaxf(r, 0.0f);
}

// ---------------------------------------------------------------------------
extern "C" void kernel_launch(void* const* d_in, const int* in_sizes, int n_in,
                              void* d_out, int out_size, void* d_ws, size_t ws_size,
                              hipStream_t stream) {
    const float*     x        = (const float*)d_in[0];
    const long long* ei       = (const long long*)d_in[1];   // int64 [2,E]
    const float*     emb      = (const float*)d_in[2];
    const float*     W        = (const float*)d_in[3];
    const float*     att_i    = (const float*)d_in[4];
    const float*     att_j    = (const float*)d_in[5];
    const float*     att_em_i = (const float*)d_in[6];
    const float*     att_em_j = (const float*)d_in[7];
    const float*     bias     = (const float*)d_in[8];
    const float*     gamma    = (const float*)d_in[9];
    const float*     beta     = (const float*)d_in[10];

    const int N = in_sizes[0] / FCH;
    const int E = in_sizes[1] / 2;
    const int Et = E + N;                 // edges + self loops

    // workspace layout (floats)
    float*    ws    = (float*)d_ws;
    float*    xl    = ws;                              // N*64
    float*    ai    = xl + (size_t)N * FCH;            // N
    float*    aj    = ai + N;                          // N
    unsigned* mmax  = (unsigned*)(aj + N);             // N
    float*    ssum  = (float*)(mmax + N);              // N
    float*    sums  = ssum + N;                        // 64
    float*    sums2 = sums + 64;                       // 64
    float*    outf  = (float*)d_out;                   // N*64 accumulator/result

    // zero-init accumulators (graph-capturable)
    hipMemsetAsync(outf, 0, (size_t)N * FCH * sizeof(float), stream);
    hipMemsetAsync(mmax, 0, (size_t)N * sizeof(unsigned), stream);  // key 0 == -inf
    hipMemsetAsync(ssum, 0, (size_t)N * sizeof(float), stream);
    hipMemsetAsync(sums, 0, 128 * sizeof(float), stream);

    // 1) xl = x @ W^T   (WMMA fp32, 32 rows x 64 cols per 256-thread block)
    gemm_xl_kernel<<<(N + 31) / 32, 256, 0, stream>>>(x, W, xl, N);

    // 2) per-node attention coefficients (one wave per node)
    node_coef_kernel<<<(N + 7) / 8, 256, 0, stream>>>(
        xl, emb, att_i, att_j, att_em_i, att_em_j, ai, aj, N);

    // 3) segment softmax over destinations
    edge_max_kernel<<<(Et + 255) / 256, 256, 0, stream>>>(ei, ai, aj, mmax, E, N);
    edge_sum_kernel<<<(Et + 255) / 256, 256, 0, stream>>>(ei, ai, aj, mmax, ssum, E, N);

    // 4) weighted scatter-add (one wave per edge)
    scatter_kernel<<<(Et + 7) / 8, 256, 0, stream>>>(
        ei, xl, ai, aj, mmax, ssum, outf, E, N);

    // 5) BatchNorm stats + apply + ReLU
    bn_stats_kernel<<<400, 256, 0, stream>>>(outf, bias, sums, sums2, N);
    bn_apply_kernel<<<(N * FCH + 255) / 256, 256, 0, stream>>>(
        outf, bias, gamma, beta, sums, sums2, N);
}